// MDRNN2D_41695542509779
// MI455X (gfx1250) — compile-verified
//
#include <hip/hip_runtime.h>
#include <math.h>

// MDRNN2D on gfx1250: 4-direction 2D GRU wavefront scan.
// f32 WMMA (dual interleaved accumulator chains), LDS-double-buffered x tiles,
// TDM (tensor_load_to_lds) for the neighbor hidden-state tile.
// B=8, C=32, H=W=128, HID=64, G=3*HID=192.

#define BSZ 8
#define CIN 32
#define HD  128
#define WD  128
#define HID 64
#define G   192
#define NTHREADS 384   // 12 wave32s; wave j owns output columns [16j, 16j+16)

typedef float v2f  __attribute__((ext_vector_type(2)));
typedef float v8f  __attribute__((ext_vector_type(8)));
typedef unsigned int u32x4 __attribute__((ext_vector_type(4)));
typedef int   i32x4 __attribute__((ext_vector_type(4)));
typedef int   i32x8 __attribute__((ext_vector_type(8)));

__global__ __launch_bounds__(NTHREADS)
void mdrnn2d_scan_kernel(const float* __restrict__ x,    // (B, C, H, W)
                         const float* __restrict__ Wx,   // (4, C, G)
                         const float* __restrict__ Wh,   // (4, HID, G)  (h_above)
                         const float* __restrict__ Wh2,  // (4, HID, G)  (h_left)
                         const float* __restrict__ bias, // (4, G)
                         float* __restrict__ out,        // [2048 final_hidden][B,HID,4,H,W map]
                         float* __restrict__ hbuf,       // ws: (4, H, W, B, HID) compact cells
                         unsigned int* __restrict__ flags) // ws: (4, W) rows-completed
{
    const int w    = blockIdx.x;          // column owned by this block
    const int d    = blockIdx.y;          // direction
    const int tid  = threadIdx.x;
    const int wave = tid >> 5;            // 0..11
    const int lane = tid & 31;
    const int m    = lane & 15;           // A-matrix row (batch idx when < 8)
    const int hi   = lane >> 4;           // lane half
    const int kb   = hi * 2;              // lanes 0-15 -> K+0/+1, lanes 16-31 -> K+2/+3
    const int col  = (wave << 4) + (lane & 15);  // output column 0..191

    // ---- Preload WMMA B-fragments for this wave's 16 output columns (register resident) ----
    const float* WxD  = Wx  + (size_t)d * CIN * G;
    const float* WhD  = Wh  + (size_t)d * HID * G;
    const float* Wh2D = Wh2 + (size_t)d * HID * G;
    v2f wxF[8], whF[16], wh2F[16];
#pragma unroll
    for (int kk = 0; kk < 8; ++kk) {
        int k = 4 * kk + kb;
        wxF[kk].x = WxD[(k + 0) * G + col];
        wxF[kk].y = WxD[(k + 1) * G + col];
    }
#pragma unroll
    for (int kk = 0; kk < 16; ++kk) {
        int k = 4 * kk + kb;
        whF[kk].x  = WhD [(k + 0) * G + col];
        whF[kk].y  = WhD [(k + 1) * G + col];
        wh2F[kk].x = Wh2D[(k + 0) * G + col];
        wh2F[kk].y = Wh2D[(k + 1) * G + col];
    }
    const float biasv = bias[d * G + col];

    // ---- LDS state ----
    __shared__ float hA[16][HID];      // h_above (rows 8..15 stay zero forever)
    __shared__ float hL[16][HID];      // h_left  (rows 0..7 = TDM dest; 8..15 zero)
    __shared__ float rg[16][G];        // per-cell preactivations
    __shared__ float xn[16][HID];      // x-part of n-gate (bias + x@Wx_n)
    __shared__ float xs[2][16][CIN];   // double-buffered per-cell x tile (rows 8..15 zero)
    for (int i = tid; i < 16 * HID; i += NTHREADS) {
        hA[i >> 6][i & 63] = 0.0f;
        hL[i >> 6][i & 63] = 0.0f;
    }
    for (int i = tid; i < 2 * 16 * CIN; i += NTHREADS)
        (&xs[0][0][0])[i] = 0.0f;
    __syncthreads();

    const int wp = (d & 2) ? (WD - 1 - w) : w;   // source/dest column in x/out space
    const unsigned int* leftFlag = flags + d * WD + (w > 0 ? (w - 1) : 0);
    unsigned int* myFlag = flags + d * WD + w;

    // Stage x for row 0.
    const int xb = tid >> 5, xc = tid & 31;      // 12 waves -> 8 loader waves used
    if (tid < BSZ * CIN) {
        const int hp0 = (d & 1) ? (HD - 1) : 0;
        xs[0][xb][xc] = x[(((size_t)xb * CIN + xc) * HD + hp0) * WD + wp];
    }
    __syncthreads();

    // ---- Constant part of the TDM descriptor: contiguous 512-f32 (2KB) tile ----
    const unsigned ldsHL = (unsigned)(unsigned long long)(const void*)&hL[0][0];
    i32x8 tdmG1;
    tdmG1[0] = (2 << 16);              // data_size = 4 bytes; no mask/iterate/pad
    tdmG1[1] = (int)(512u << 16);      // tensor_dim0 = 512  (bits 63:48)
    tdmG1[2] = (int)(1u << 16);        // tensor_dim1 = 1    (bits 95:80)
    tdmG1[3] = (int)(512u << 16);      // tile_dim0 = 512    (bits 127:112)
    tdmG1[4] = 1;                      // tile_dim1 = 1      (bits 143:128)
    tdmG1[5] = 512;                    // tensor_dim0_stride = 512
    tdmG1[6] = 0;
    tdmG1[7] = 0;
    const i32x4 tdmZ4 = {0, 0, 0, 0};

    for (int h = 0; h < HD; ++h) {
        const int hp  = (d & 1) ? (HD - 1 - h) : h;  // source/dest row in x/out space
        const int cur = h & 1, nxt = cur ^ 1;

        // -------- A. issue next row's x loads (held in a register until late) --------
        float xval = 0.0f;
        const bool doNext = (h + 1 < HD) && (tid < BSZ * CIN);
        if (doNext) {
            const int hp1 = (d & 1) ? (HD - 2 - h) : (h + 1);
            xval = x[(((size_t)xb * CIN + xc) * HD + hp1) * WD + wp];
        }

        // -------- B. input projection: bias + x_cell(8x32) @ Wx(32x192), dual chains --------
        v8f accA, accB;
#pragma unroll
        for (int v = 0; v < 8; ++v) { accA[v] = biasv; accB[v] = 0.0f; }
#pragma unroll
        for (int kk = 0; kk < 8; kk += 2) {
            v2f a0, a1;
            a0.x = xs[cur][m][4 * kk + kb];     a0.y = xs[cur][m][4 * kk + kb + 1];
            a1.x = xs[cur][m][4 * kk + 4 + kb]; a1.y = xs[cur][m][4 * kk + 4 + kb + 1];
            accA = __builtin_amdgcn_wmma_f32_16x16x4_f32(false, a0, false, wxF[kk],
                                                         (short)0, accA, false, false);
            accB = __builtin_amdgcn_wmma_f32_16x16x4_f32(false, a1, false, wxF[kk + 1],
                                                         (short)0, accB, false, false);
        }

        // n-gate waves keep x-part separate (n = tanh(xn + r*hn)); restart accumulators
        if (wave >= 8) {
            const int nc = col - 128;
#pragma unroll
            for (int v = 0; v < 8; ++v) xn[v + 8 * hi][nc] = accA[v] + accB[v];
#pragma unroll
            for (int v = 0; v < 8; ++v) { accA[v] = 0.0f; accB[v] = 0.0f; }
        }

        // -------- C. acquire h_left tile from left neighbor via the Tensor Data Mover --------
        if (w > 0) {
            if (tid == 0) {
                while (__hip_atomic_load(leftFlag, __ATOMIC_ACQUIRE,
                                         __HIP_MEMORY_SCOPE_AGENT) <= (unsigned)h) {
                    __builtin_amdgcn_s_sleep(1);
                }
            }
            __syncthreads();
            if (wave == 0) {
                const unsigned long long ga = (unsigned long long)(const void*)
                    (hbuf + ((((size_t)d * HD + h) * WD + (w - 1)) * (BSZ * HID)));
                u32x4 g0;
                g0[0] = 1u;                                   // count=1, no gather
                g0[1] = ldsHL;                                // LDS dest byte offset
                g0[2] = (unsigned)ga;                         // global_addr[31:0]
                g0[3] = ((unsigned)(ga >> 32) & 0x01FFFFFFu)  // global_addr[56:32]
                        | (2u << 30);                         // type = 2 (image/tensor)
#if __clang_major__ >= 23
                const i32x8 z8 = {0, 0, 0, 0, 0, 0, 0, 0};
                __builtin_amdgcn_tensor_load_to_lds(g0, tdmG1, tdmZ4, tdmZ4, z8, 0);
#else
                __builtin_amdgcn_tensor_load_to_lds(g0, tdmG1, tdmZ4, tdmZ4, 0);
#endif
                __builtin_amdgcn_s_wait_tensorcnt((short)0);
            }
        }
        __syncthreads();  // hL / xn ready, previous combine fully consumed

        // -------- D. recurrence GEMMs: hA@Wh (chain A) interleaved with hL@Wh2 (chain B) ----
#pragma unroll
        for (int kk = 0; kk < 16; ++kk) {
            const int k = 4 * kk + kb;
            v2f aA, aL;
            aA.x = hA[m][k]; aA.y = hA[m][k + 1];
            aL.x = hL[m][k]; aL.y = hL[m][k + 1];
            accA = __builtin_amdgcn_wmma_f32_16x16x4_f32(false, aA, false, whF[kk],
                                                         (short)0, accA, false, false);
            accB = __builtin_amdgcn_wmma_f32_16x16x4_f32(false, aL, false, wh2F[kk],
                                                         (short)0, accB, false, false);
        }
#pragma unroll
        for (int v = 0; v < 8; ++v) rg[v + 8 * hi][col] = accA[v] + accB[v];

        // Commit next row's x tile to LDS (consumed only after the end-of-iter barrier).
        if (doNext) xs[nxt][xb][xc] = xval;
        __syncthreads();

        // -------- E. gate nonlinearities + state update --------
        float* cellDst = hbuf + ((((size_t)d * HD + h) * WD + w) * (BSZ * HID));
        for (int i = tid; i < BSZ * HID; i += NTHREADS) {
            const int b = i >> 6, q = i & 63;
            const float z = 1.0f / (1.0f + __expf(-rg[b][q]));
            const float r = 1.0f / (1.0f + __expf(-rg[b][64 + q]));
            const float n = tanhf(xn[b][q] + r * rg[b][128 + q]);
            const float hnew = (1.0f - z) * n + z * 0.5f * (hA[b][q] + hL[b][q]);
            hA[b][q] = hnew;                       // becomes h_above for next row
            cellDst[i] = hnew;                     // compact cell for right neighbor's TDM
            // un-flipped output map: out_map[b][q][d][hp][wp]
            out[2048 + ((((size_t)b * HID + q) * 4 + d) * HD + hp) * WD + wp] = hnew;
            if (h == HD - 1 && w == WD - 1)        // final hidden: concat over d
                out[(size_t)b * (4 * HID) + d * HID + q] = hnew;
        }
        __threadfence();   // release our row's stores to agent scope
        __syncthreads();
        if (tid == 0)
            __hip_atomic_store(myFlag, (unsigned)(h + 1), __ATOMIC_RELEASE,
                               __HIP_MEMORY_SCOPE_AGENT);
    }
}

extern "C" void kernel_launch(void* const* d_in, const int* in_sizes, int n_in,
                              void* d_out, int out_size, void* d_ws, size_t ws_size,
                              hipStream_t stream) {
    const float* x    = (const float*)d_in[0];
    const float* Wx   = (const float*)d_in[1];
    const float* Wh   = (const float*)d_in[2];
    const float* Wh2  = (const float*)d_in[3];
    const float* bias = (const float*)d_in[4];
    float* out = (float*)d_out;

    const size_t hbuf_bytes = (size_t)4 * HD * WD * BSZ * HID * sizeof(float); // 134.2 MB
    float* hbuf = (float*)d_ws;
    unsigned int* flags = (unsigned int*)((char*)d_ws + hbuf_bytes);

    hipMemsetAsync(flags, 0, (size_t)4 * WD * sizeof(unsigned int), stream);

    dim3 grid(WD, 4);   // 128 columns x 4 directions = 512 pipelined blocks
    mdrnn2d_scan_kernel<<<grid, NTHREADS, 0, stream>>>(x, Wx, Wh, Wh2, bias,
                                                       out, hbuf, flags);
}